// CodeCloud_58987080843720
// MI455X (gfx1250) — compile-verified
//
#include <hip/hip_runtime.h>

typedef float v2f __attribute__((ext_vector_type(2)));
typedef float v8f __attribute__((ext_vector_type(8)));

#define BATCH   32
#define NPTS    4096
#define NCODES  128
#define CDIM    128
#define PT      128        // points per workgroup
#define STR     130        // padded LDS row stride (even -> b64 aligned, odd/2 -> no bank conflicts)
#define THREADS 256
#define EPSF    1e-16f

__global__ __launch_bounds__(THREADS)
void codecloud_wmma_kernel(const int*   __restrict__ indices,
                           const float* __restrict__ qpts,        // (B,P,3)
                           const float* __restrict__ cpos,        // (R,C,3)
                           const float* __restrict__ codes,       // (R,C,D)
                           const int*   __restrict__ dist_scale,  // scalar
                           float* __restrict__ out_qc,            // (B,P,D)
                           float* __restrict__ out_sqd,           // (B,P,C)
                           float* __restrict__ out_wt)            // (B,P,C)
{
    extern __shared__ float smem[];
    float* Wt   = smem;                    // [PT][STR]   weights (unnorm -> norm)
    float* Sb   = smem + PT * STR;         // [C][STR]    phase2/3: sqd tile [PT][C]; phase4/5: codes^T [D][C]
    float* px   = Sb + NCODES * STR;
    float* py   = px + NCODES;
    float* pz   = py + NCODES;
    float* rinv = pz + NCODES;             // [PT] reciprocal row sums

    const int tid  = threadIdx.x;
    const int lane = tid & 31;
    const int wave = tid >> 5;
    const int b    = blockIdx.y;
    const int p0   = blockIdx.x * PT;
    const int rec  = indices[b];
    const int s    = dist_scale[0];

    // ---- phase 0: warm L2/WGP$ with this record's 64KB code tile (global_prefetch_b8).
    // No counters, no VGPR returns; overlaps the tile fetch with the distance math below.
    {
        const char* cbase = (const char*)(codes + (size_t)rec * NCODES * CDIM);
        __builtin_prefetch(cbase + (size_t)tid * 256,       0, 1);
        __builtin_prefetch(cbase + (size_t)tid * 256 + 128, 0, 1);
    }

    // ---- phase 1: code positions SoA into LDS ----
    if (tid < NCODES) {
        const float* pp = cpos + ((size_t)rec * NCODES + tid) * 3;
        px[tid] = pp[0]; py[tid] = pp[1]; pz[tid] = pp[2];
    }
    __syncthreads();

    // ---- phase 2: distances + unnormalized weights (2 threads per point) ----
    {
        const int p  = tid >> 1;
        const int c0 = (tid & 1) * (NCODES / 2);
        const float* q = qpts + ((size_t)b * NPTS + p0 + p) * 3;
        const float qx = q[0], qy = q[1], qz = q[2];
        float ssum = 0.0f;
        #pragma unroll 4
        for (int i = 0; i < NCODES / 2; ++i) {
            const int c = c0 + i;
            const float dx = qx - px[c];
            const float dy = qy - py[c];
            const float dz = qz - pz[c];
            const float d  = fmaf(dx, dx, fmaf(dy, dy, dz * dz)) + EPSF;
            Sb[p * NCODES + c] = d;                          // stage square_dist
            const float w = (s == 2) ? (1.0f / d)            // fast path: 1/sqrt(d)^2
                                     : __powf(d, -0.5f * (float)s);
            Wt[p * STR + c] = w;
            ssum += w;
        }
        const float tot = ssum + __shfl_xor(ssum, 1);        // lane pair shares one point
        if ((tid & 1) == 0) rinv[p] = 1.0f / tot;
    }
    __syncthreads();

    // ---- phase 3: coalesced square_dist / normalized-weight writeback ----
    for (int idx = tid; idx < PT * NCODES; idx += THREADS) {
        const int p = idx >> 7;                // /NCODES
        const int c = idx & (NCODES - 1);
        const size_t g = ((size_t)b * NPTS + p0 + p) * NCODES + c;
        out_sqd[g] = Sb[idx];
        const float wn = Wt[p * STR + c] * rinv[p];
        Wt[p * STR + c] = wn;
        out_wt[g] = wn;
    }
    __syncthreads();

    // ---- phase 4: stage codes^T into LDS: Sb[d][c] = codes[rec][c][d] ----
    for (int idx = tid; idx < NCODES * (CDIM / 4); idx += THREADS) {
        const int c  = idx >> 5;               // CDIM/4 == 32
        const int d0 = (idx & 31) << 2;
        const float4 v = *(const float4*)(codes + ((size_t)rec * NCODES + c) * CDIM + d0);
        Sb[(d0 + 0) * STR + c] = v.x;
        Sb[(d0 + 1) * STR + c] = v.y;
        Sb[(d0 + 2) * STR + c] = v.z;
        Sb[(d0 + 3) * STR + c] = v.w;
    }
    __syncthreads();

    // ---- phase 5: (PT x C) @ (C x D) via V_WMMA_F32_16X16X4_F32 ----
    // A 16x4 f32 layout: lanes 0-15 -> K=0,1 ; lanes 16-31 -> K=2,3 (ISA 7.12.2); B mirrored.
    // Each wave has fixed nt (= wave): compiler keeps all 32 B-fragments register-resident.
    const int half  = lane >> 4;               // 0 or 1
    const int kbase = half << 1;               // K offset 0 or 2
    const int mn    = lane & 15;
    for (int t = wave; t < (PT / 16) * (CDIM / 16); t += THREADS / 32) {
        const int mt = t >> 3;
        const int nt = t & 7;
        const float* Arow = Wt + (mt * 16 + mn) * STR + kbase;   // weight row (M x K)
        const float* Brow = Sb + (nt * 16 + mn) * STR + kbase;   // codes^T row (N x K)
        v8f acc = {};
        #pragma unroll
        for (int k = 0; k < NCODES; k += 4) {
            v2f a  = *(const v2f*)(Arow + k);                    // ds_load_2addr_b64, conflict-free
            v2f bb = *(const v2f*)(Brow + k);
            acc = __builtin_amdgcn_wmma_f32_16x16x4_f32(
                      /*neg_a=*/false, a, /*neg_b=*/false, bb,
                      /*c_mod=*/(short)0, acc, /*reuse_a=*/false, /*reuse_b=*/false);
        }
        // C/D layout: VGPR v -> M = mt*16 + v + (lane>=16 ? 8 : 0), N = nt*16 + (lane&15)
        const int col  = nt * 16 + mn;
        const int row0 = mt * 16 + (half << 3);
        #pragma unroll
        for (int v = 0; v < 8; ++v) {
            out_qc[((size_t)b * NPTS + p0 + row0 + v) * CDIM + col] = acc[v];
        }
    }
}

extern "C" void kernel_launch(void* const* d_in, const int* in_sizes, int n_in,
                              void* d_out, int out_size, void* d_ws, size_t ws_size,
                              hipStream_t stream) {
    const int*   indices = (const int*)d_in[0];
    const float* qpts    = (const float*)d_in[1];
    const float* cpos    = (const float*)d_in[2];
    const float* codes   = (const float*)d_in[3];
    const int*   dscale  = (const int*)d_in[4];

    float* out     = (float*)d_out;
    float* out_qc  = out;                                          // (B,P,D)
    float* out_sqd = out + (size_t)BATCH * NPTS * CDIM;            // (B,P,C)
    float* out_wt  = out_sqd + (size_t)BATCH * NPTS * NCODES;      // (B,P,C)

    const size_t shmem = (size_t)(PT * STR + NCODES * STR + 3 * NCODES + PT) * sizeof(float);
    dim3 grid(NPTS / PT, BATCH);
    codecloud_wmma_kernel<<<grid, THREADS, shmem, stream>>>(
        indices, qpts, cpos, codes, dscale, out_qc, out_sqd, out_wt);
}